// DLTKcat_20169166422462
// MI455X (gfx1250) — compile-verified
//
#include <hip/hip_runtime.h>
#include <math.h>

// ---------------- model dims ----------------
constexpr int Bq   = 32;    // batch
constexpr int Nq   = 128;   // atoms
constexpr int Mq   = 1024;  // aminos
constexpr int CDq  = 128;
constexpr int GDq  = 64;
constexpr int NHq  = 4;
constexpr int LDq  = 256;
constexpr int PDq  = 40;
constexpr int KWq  = 11;    // conv kernel (2*WINDOW+1)
constexpr int VD   = 3 * LDq + 2; // 770

typedef __attribute__((ext_vector_type(2))) float v2f;
typedef __attribute__((ext_vector_type(8))) float v8f;

// explicit global-address-space pointers so the backend emits
// global_load_* / global_store_* (not flat_*: flat dual-counts
// LOADcnt+DScnt and ties up the LDS datapath)
typedef __attribute__((address_space(1))) const float* gcfp;
typedef __attribute__((address_space(1)))       float* gfp;
typedef __attribute__((address_space(1))) const v2f*   gv2fp;

// =====================================================================
// Batched WMMA GEMM:  C = act( A(*) @ B(*) + bias ) * rowS * colS
// One wave computes a 64x64 tile as 4x4 WMMA 16x16 subtiles.
// No divergent branches in the K loop: out-of-range rows/cols are
// address-clamped (their results are never stored); only the K tail
// (K % 4 != 0) zero-masks operands.
// =====================================================================
template<int TA, int TB>
__global__ __launch_bounds__(32)
void gemm_wmma_kernel(const float* __restrict__ A, long long aB1, long long aB2, int lda,
                      const float* __restrict__ Bm, long long bB1, long long bB2, int ldb,
                      float* __restrict__ C, long long cB1, long long cB2, int ldc,
                      const float* __restrict__ bias,
                      const float* __restrict__ rowScale, long long rsB,
                      const float* __restrict__ colScale, long long csB,
                      int Mt, int Nt, int Kt, int batchDiv, int act)
{
    const int z  = blockIdx.z;
    const int z1 = z / batchDiv;
    const int z2 = z - z1 * batchDiv;
    gcfp Ab = (gcfp)A  + (long long)z1 * aB1 + (long long)z2 * aB2;
    gcfp Bb = (gcfp)Bm + (long long)z1 * bB1 + (long long)z2 * bB2;
    gfp  Cb = (gfp)C   + (long long)z1 * cB1 + (long long)z2 * cB2;
    gcfp biasg = (gcfp)bias;
    gcfp rowSg = (gcfp)rowScale;
    gcfp colSg = (gcfp)colScale;

    const int lane = threadIdx.x;
    const int half = lane >> 4;          // K-pair selector per ISA A/B layout
    const int r    = lane & 15;
    const int m0   = blockIdx.y * 64;
    const int n0   = blockIdx.x * 64;

    int arow[4], bcol[4];
#pragma unroll
    for (int t = 0; t < 4; ++t) {
        arow[t] = min(m0 + 16 * t + r, Mt - 1);   // clamp: never stored if OOR
        bcol[t] = min(n0 + 16 * t + r, Nt - 1);
    }

    v8f acc[4][4];
#pragma unroll
    for (int mi = 0; mi < 4; ++mi)
#pragma unroll
        for (int ni = 0; ni < 4; ++ni)
            acc[mi][ni] = (v8f){};

    const int ks0 = 2 * half;
    gcfp pA[4];
    gcfp pB[4];
#pragma unroll
    for (int t = 0; t < 4; ++t) {
        pA[t] = TA ? (Ab + (long long)ks0 * lda + arow[t])
                   : (Ab + (long long)arow[t] * lda + ks0);
        pB[t] = TB ? (Bb + (long long)bcol[t] * ldb + ks0)
                   : (Bb + (long long)ks0 * ldb + bcol[t]);
    }
    const long long stepA = TA ? (long long)4 * lda : 4;
    const long long stepB = TB ? 4 : (long long)4 * ldb;

    const int K4 = Kt & ~3;
    for (int k = 0; k < K4; k += 4) {
        v2f a[4], b[4];
#pragma unroll
        for (int t = 0; t < 4; ++t) {
            if (TA) { a[t].x = pA[t][0]; a[t].y = pA[t][lda]; }
            else    { a[t] = *(gv2fp)pA[t]; }             // contiguous K: b64 load
            pA[t] += stepA;
        }
#pragma unroll
        for (int t = 0; t < 4; ++t) {
            if (TB) { b[t] = *(gv2fp)pB[t]; }             // contiguous K: b64 load
            else    { b[t].x = pB[t][0]; b[t].y = pB[t][ldb]; }
            pB[t] += stepB;
        }
#pragma unroll
        for (int mi = 0; mi < 4; ++mi)
#pragma unroll
            for (int ni = 0; ni < 4; ++ni)
                acc[mi][ni] = __builtin_amdgcn_wmma_f32_16x16x4_f32(
                    false, a[mi], false, b[ni], (short)0, acc[mi][ni], false, false);
    }

    if (K4 < Kt) {   // K tail (only the 770-wide MLP GEMMs hit this)
        const int ks = K4 + ks0;
        const int k0 = min(ks,     Kt - 1);
        const int k1 = min(ks + 1, Kt - 1);
        const float f0 = (ks     < Kt) ? 1.f : 0.f;
        const float f1 = (ks + 1 < Kt) ? 1.f : 0.f;
        v2f a[4], b[4];
#pragma unroll
        for (int t = 0; t < 4; ++t) {
            float ax = TA ? Ab[(long long)k0 * lda + arow[t]] : Ab[(long long)arow[t] * lda + k0];
            float ay = TA ? Ab[(long long)k1 * lda + arow[t]] : Ab[(long long)arow[t] * lda + k1];
            a[t].x = ax * f0; a[t].y = ay * f1;           // zero A side kills OOR products
            b[t].x = TB ? Bb[(long long)bcol[t] * ldb + k0] : Bb[(long long)k0 * ldb + bcol[t]];
            b[t].y = TB ? Bb[(long long)bcol[t] * ldb + k1] : Bb[(long long)k1 * ldb + bcol[t]];
        }
#pragma unroll
        for (int mi = 0; mi < 4; ++mi)
#pragma unroll
            for (int ni = 0; ni < 4; ++ni)
                acc[mi][ni] = __builtin_amdgcn_wmma_f32_16x16x4_f32(
                    false, a[mi], false, b[ni], (short)0, acc[mi][ni], false, false);
    }

    // -------- epilogue --------
#pragma unroll
    for (int ni = 0; ni < 4; ++ni) {
        const int  col = n0 + 16 * ni + r;
        const bool cok = col < Nt;
        const float bval = bias     ? biasg[bcol[ni]] : 0.f;                      // clamped idx
        const float cs   = colScale ? colSg[(long long)z1 * csB + bcol[ni]] : 1.f;
#pragma unroll
        for (int mi = 0; mi < 4; ++mi) {
#pragma unroll
            for (int v = 0; v < 8; ++v) {
                const int row = m0 + 16 * mi + v + 8 * half;
                if (cok && row < Mt) {
                    float x = acc[mi][ni][v] + bval;
                    if      (act == 1) x = (x > 0.f) ? x : 0.2f * x;   // leaky_relu
                    else if (act == 2) x = (x > 0.f) ? x : expm1f(x);  // elu
                    else if (act == 3) x = tanhf(x);                   // tanh
                    if (rowScale) x *= rowSg[(long long)z1 * rsB + row];
                    x *= cs;
                    Cb[(long long)row * ldc + col] = x;
                }
            }
        }
    }
}

// ---- repack gat_W (NH,CD,GD) -> Wcat (CD, NH*GD) head-major columns ----
__global__ void pack_wcat_kernel(const float* __restrict__ gw, float* __restrict__ wcat)
{
    int idx = blockIdx.x * blockDim.x + threadIdx.x;
    if (idx >= CDq * NHq * GDq) return;
    int col = idx % (NHq * GDq);
    int k   = idx / (NHq * GDq);
    int i   = col >> 6;          // head
    int g   = col & 63;
    wcat[idx] = gw[((long long)i * CDq + k) * GDq + g];
}

// ---- s1[b,i,n] = Wh . a[:G], s2[b,i,n] = Wh . a[G:2G] ----
__global__ void gat_scores_kernel(const float* __restrict__ wh, int ldwh,
                                  const float* __restrict__ avec, int aStride,
                                  float* __restrict__ s1, float* __restrict__ s2,
                                  int NH, int G)
{
    int idx = blockIdx.x * blockDim.x + threadIdx.x;
    int total = Bq * NH * Nq;
    if (idx >= total) return;
    int n = idx % Nq;
    int t = idx / Nq;
    int i = t % NH;
    int b = t / NH;
    const float* w  = wh + ((long long)(b * Nq + n)) * ldwh + i * G;
    const float* a1 = avec + (long long)i * aStride;
    const float* a2 = a1 + G;
    float x1 = 0.f, x2 = 0.f;
    for (int g = 0; g < G; ++g) { float v = w[g]; x1 += v * a1[g]; x2 += v * a2[g]; }
    s1[idx] = x1; s2[idx] = x2;
}

// ---- GAT masked softmax attention: att[b,i,n,m] over m ----
__global__ __launch_bounds__(128)
void gat_att_kernel(const float* __restrict__ s1, const float* __restrict__ s2,
                    const int* __restrict__ adj, float* __restrict__ att, int NH)
{
    int n = blockIdx.x, i = blockIdx.y, b = blockIdx.z, m = threadIdx.x;
    long long sbase = ((long long)(b * NH + i)) * Nq;
    float e = s1[sbase + n] + s2[sbase + m];
    e = (e > 0.f) ? e : 0.2f * e;
    int ad = adj[((long long)b * Nq + n) * Nq + m];
    if (ad <= 0) e = -9.0e15f;
    __shared__ float red[128];
    red[m] = e; __syncthreads();
    for (int s = 64; s > 0; s >>= 1) { if (m < s) red[m] = fmaxf(red[m], red[m + s]); __syncthreads(); }
    float mx = red[0]; __syncthreads();
    float ex = __expf(e - mx);
    red[m] = ex; __syncthreads();
    for (int s = 64; s > 0; s >>= 1) { if (m < s) red[m] += red[m + s]; __syncthreads(); }
    att[(sbase + n) * Nq + m] = ex / red[0];
}

// ---- 11x11 same-padded 2D conv over (M,PD) plane, leaky_relu epilogue ----
__global__ void conv2d_kernel(const float* __restrict__ in, const float* __restrict__ w,
                              const float* __restrict__ bias, float* __restrict__ out)
{
    int idx = blockIdx.x * blockDim.x + threadIdx.x;
    int total = Bq * Mq * PDq;
    if (idx >= total) return;
    int c = idx % PDq;
    int y = (idx / PDq) % Mq;
    int b = idx / (PDq * Mq);
    float s = bias[0];
    for (int ky = 0; ky < KWq; ++ky) {
        int iy = y + ky - 5;
        if (iy < 0 || iy >= Mq) continue;
        const float* row = in + ((long long)b * Mq + iy) * PDq;
        for (int kx = 0; kx < KWq; ++kx) {
            int ix = c + kx - 5;
            if (ix < 0 || ix >= PDq) continue;
            s += row[ix] * w[ky * KWq + kx];
        }
    }
    out[idx] = (s > 0.f) ? s : 0.2f * s;
}

// ---- atom-side attention score + mask softmax over N=128 ----
__global__ __launch_bounds__(128)
void atoms_att_kernel(const float* __restrict__ bhc, const float* __restrict__ atr,
                      const float* __restrict__ wv, const float* __restrict__ bsc,
                      const float* __restrict__ mask, float* __restrict__ out)
{
    int b = blockIdx.x, n = threadIdx.x;
    const float* p1 = bhc + ((long long)(b * Nq + n)) * LDq;
    const float* p2 = atr + ((long long)(b * Nq + n)) * LDq;
    float s = bsc[0];
    for (int d = 0; d < LDq; ++d) s += p1[d] * wv[d];
    for (int d = 0; d < LDq; ++d) s += p2[d] * wv[LDq + d];
    __shared__ float red[128];
    red[n] = s; __syncthreads();
    for (int st = 64; st > 0; st >>= 1) { if (n < st) red[n] = fmaxf(red[n], red[n + st]); __syncthreads(); }
    float mx = red[0]; __syncthreads();
    float ex = __expf(s - mx) * mask[b * Nq + n];
    red[n] = ex; __syncthreads();
    for (int st = 64; st > 0; st >>= 1) { if (n < st) red[n] += red[n + st]; __syncthreads(); }
    out[b * Nq + n] = ex / (red[0] + 1e-6f);
}

// ---- amino-side attention score + mask softmax over M=1024 ----
__global__ __launch_bounds__(1024)
void amino_att_kernel(const float* __restrict__ bhp, const float* __restrict__ amt,
                      const float* __restrict__ wv, const float* __restrict__ bsc,
                      const float* __restrict__ mask, float* __restrict__ out)
{
    int b = blockIdx.x, m = threadIdx.x;
    const float* p1 = bhp + ((long long)(b * Mq + m)) * LDq;
    const float* p2 = amt + ((long long)(b * Mq + m)) * LDq;
    float s = bsc[0];
    for (int d = 0; d < LDq; ++d) s += p1[d] * wv[d];
    for (int d = 0; d < LDq; ++d) s += p2[d] * wv[LDq + d];
    __shared__ float red[1024];
    red[m] = s; __syncthreads();
    for (int st = 512; st > 0; st >>= 1) { if (m < st) red[m] = fmaxf(red[m], red[m + st]); __syncthreads(); }
    float mx = red[0]; __syncthreads();
    float ex = __expf(s - mx) * mask[b * Mq + m];
    red[m] = ex; __syncthreads();
    for (int st = 512; st > 0; st >>= 1) { if (m < st) red[m] += red[m + st]; __syncthreads(); }
    out[b * Mq + m] = ex / (red[0] + 1e-6f);
}

// ---- attention-weighted pooling: out[b, colOff+d] = sum_n feat[b,n,d]*att[b,n] ----
__global__ __launch_bounds__(256)
void pool_kernel(const float* __restrict__ feat, const float* __restrict__ att,
                 float* __restrict__ out, int Nn, int colOff, int ldout)
{
    int b = blockIdx.x, d = threadIdx.x;
    float s = 0.f;
    for (int n = 0; n < Nn; ++n)
        s += feat[((long long)(b * Nn + n)) * LDq + d] * att[b * Nn + n];
    out[(long long)b * ldout + colOff + d] = s;
}

// ---- v = concat([cf_lin, fps, pf_final, inv_Temp, Temp]) ----
__global__ __launch_bounds__(800)
void build_v_kernel(const float* __restrict__ cf, const float* __restrict__ pf,
                    const float* __restrict__ fps, const float* __restrict__ invT,
                    const float* __restrict__ T, float* __restrict__ v)
{
    int b = blockIdx.x, j = threadIdx.x;
    if (j >= VD) return;
    float x;
    if      (j < 256) x = cf[b * 256 + j];
    else if (j < 512) x = fps[b * 256 + (j - 256)];
    else if (j < 768) x = pf[b * 256 + (j - 512)];
    else if (j == 768) x = invT[b];
    else               x = T[b];
    v[(long long)b * VD + j] = x;
}

// ---- final scalar head ----
__global__ void out_kernel(const float* __restrict__ v, const float* __restrict__ w,
                           const float* __restrict__ bptr, float* __restrict__ out)
{
    int b = blockIdx.x * blockDim.x + threadIdx.x;
    if (b >= Bq) return;
    float s = bptr[0];
    for (int j = 0; j < VD; ++j) s += v[(long long)b * VD + j] * w[j];
    out[b] = s;
}

// =====================================================================
static void gemm(hipStream_t st,
                 const float* A, long long aB1, long long aB2, int lda, int transA,
                 const float* Bm, long long bB1, long long bB2, int ldb, int transB,
                 float* C, long long cB1, long long cB2, int ldc,
                 const float* bias,
                 const float* rowScale, long long rsB,
                 const float* colScale, long long csB,
                 int Mt, int Nt, int Kt, int batch, int batchDiv, int act)
{
    dim3 g((Nt + 63) / 64, (Mt + 63) / 64, batch);
    if (!transA && !transB)
        gemm_wmma_kernel<0,0><<<g, 32, 0, st>>>(A,aB1,aB2,lda, Bm,bB1,bB2,ldb, C,cB1,cB2,ldc,
                                                bias, rowScale,rsB, colScale,csB, Mt,Nt,Kt,batchDiv,act);
    else if (!transA && transB)
        gemm_wmma_kernel<0,1><<<g, 32, 0, st>>>(A,aB1,aB2,lda, Bm,bB1,bB2,ldb, C,cB1,cB2,ldc,
                                                bias, rowScale,rsB, colScale,csB, Mt,Nt,Kt,batchDiv,act);
    else if (transA && !transB)
        gemm_wmma_kernel<1,0><<<g, 32, 0, st>>>(A,aB1,aB2,lda, Bm,bB1,bB2,ldb, C,cB1,cB2,ldc,
                                                bias, rowScale,rsB, colScale,csB, Mt,Nt,Kt,batchDiv,act);
    else
        gemm_wmma_kernel<1,1><<<g, 32, 0, st>>>(A,aB1,aB2,lda, Bm,bB1,bB2,ldb, C,cB1,cB2,ldc,
                                                bias, rowScale,rsB, colScale,csB, Mt,Nt,Kt,batchDiv,act);
}

extern "C" void kernel_launch(void* const* d_in, const int* in_sizes, int n_in,
                              void* d_out, int out_size, void* d_ws, size_t ws_size,
                              hipStream_t stream)
{
    (void)in_sizes; (void)n_in; (void)out_size; (void)ws_size;
    const float* atoms_emb = (const float*)d_in[0];
    const int*   adjacency = (const int*)  d_in[1];
    const float* atoms_mask= (const float*)d_in[2];
    const float* amino_emb = (const float*)d_in[3];
    const float* amino_mask= (const float*)d_in[4];
    const float* fps       = (const float*)d_in[5];
    const float* inv_Temp  = (const float*)d_in[6];
    const float* Temp      = (const float*)d_in[7];
    const float* bert_W    = (const float*)d_in[8];
    const float* bert_b    = (const float*)d_in[9];
    const float* gat_W     = (const float*)d_in[10];
    const float* gat_a     = (const float*)d_in[11];
    const float* gatout_W  = (const float*)d_in[12];
    const float* gatout_a  = (const float*)d_in[13];
    const float* Wcomp_W   = (const float*)d_in[14];
    const float* Wcomp_b   = (const float*)d_in[15];
    const float* prot_W    = (const float*)d_in[16];
    const float* prot_b    = (const float*)d_in[17];
    const float* conv_W    = (const float*)d_in[18];
    const float* conv_b    = (const float*)d_in[19];
    const float* Wprot_W   = (const float*)d_in[20];
    const float* Wprot_b   = (const float*)d_in[21];
    const float* Uw        = (const float*)d_in[22];
    const float* tc2p_W    = (const float*)d_in[23];
    const float* tc2p_b    = (const float*)d_in[24];
    const float* tp2c_W    = (const float*)d_in[25];
    const float* tp2c_b    = (const float*)d_in[26];
    const float* bhc_W     = (const float*)d_in[27];
    const float* bhc_b     = (const float*)d_in[28];
    const float* bhp_W     = (const float*)d_in[29];
    const float* bhp_b     = (const float*)d_in[30];
    const float* battc_W   = (const float*)d_in[31];
    const float* battc_b   = (const float*)d_in[32];
    const float* battp_W   = (const float*)d_in[33];
    const float* battp_b   = (const float*)d_in[34];
    const float* combc_W   = (const float*)d_in[35];
    const float* combc_b   = (const float*)d_in[36];
    const float* combp_W   = (const float*)d_in[37];
    const float* combp_b   = (const float*)d_in[38];
    const float* Wout_W    = (const float*)d_in[39];
    const float* Wout_b    = (const float*)d_in[40];
    const float* out_W     = (const float*)d_in[41];
    const float* out_b     = (const float*)d_in[42];
    float* out = (float*)d_out;
    float* ws  = (float*)d_ws;

    // -------- workspace layout (floats) --------
    size_t o = 0;
    auto alloc = [&](size_t n) { size_t r = o; o += n; return r; };
    float* H     = ws + alloc((size_t)Bq * Nq * CDq);        // 4096x128
    float* WCAT  = ws + alloc((size_t)CDq * NHq * GDq);      // 128x256
    float* WHALL = ws + alloc((size_t)Bq * Nq * NHq * GDq);  // 4096x256
    float* SB1   = ws + alloc((size_t)Bq * NHq * Nq);
    float* SB2   = ws + alloc((size_t)Bq * NHq * Nq);
    float* ATTB  = ws + alloc((size_t)Bq * NHq * Nq * Nq);   // 8MB
    float* MULTI = ws + alloc((size_t)Bq * Nq * NHq * GDq);
    float* WH2   = ws + alloc((size_t)Bq * Nq * CDq);
    float* SB1o  = ws + alloc((size_t)Bq * Nq);
    float* SB2o  = ws + alloc((size_t)Bq * Nq);
    float* ATT2  = ws + alloc((size_t)Bq * Nq * Nq);
    float* HP2   = ws + alloc((size_t)Bq * Nq * CDq);
    float* AV    = ws + alloc((size_t)Bq * Nq * LDq);        // atoms features
    float* CA    = ws + alloc((size_t)Bq * Mq * PDq);        // conv ping
    float* CB    = ws + alloc((size_t)Bq * Mq * PDq);        // conv pong
    float* PV    = ws + alloc((size_t)Bq * Mq * LDq);        // amino features
    float* TPC   = ws + alloc((size_t)Bq * Mq * LDq);
    float* TCP   = ws + alloc((size_t)Bq * Nq * LDq);
    float* BHC   = ws + alloc((size_t)Bq * Nq * LDq);
    float* BHP   = ws + alloc((size_t)Bq * Mq * LDq);
    float* AVU   = ws + alloc((size_t)Bq * Nq * LDq);
    float* ABUF  = ws + alloc((size_t)Bq * Nq * Mq);         // interaction A
    float* ATR   = ws + alloc((size_t)Bq * Nq * LDq);        // atoms_trans
    float* AMT   = ws + alloc((size_t)Bq * Mq * LDq);        // amino_trans
    float* SCC   = ws + alloc((size_t)Bq * Nq);
    float* SCP   = ws + alloc((size_t)Bq * Mq);
    float* CATCF = ws + alloc((size_t)Bq * NHq * LDq);       // 32x1024
    float* CATPF = ws + alloc((size_t)Bq * NHq * LDq);
    float* CF    = ws + alloc((size_t)Bq * LDq);
    float* PF    = ws + alloc((size_t)Bq * LDq);
    float* V0    = ws + alloc((size_t)Bq * VD);
    float* V1    = ws + alloc((size_t)Bq * VD);

    const long long NN = (long long)Nq * Nq;

    // ---- 1) h = atoms_emb @ bert_W + bert_b :: (4096,128) K=300 ----
    gemm(stream, atoms_emb, 0,0, 300,0, bert_W, 0,0, CDq,0, H, 0,0, CDq,
         bert_b, nullptr,0, nullptr,0, Bq*Nq, CDq, 300, 1,1, 0);

    // ---- 2) Wh for all heads (head-major columns == multi layout) ----
    pack_wcat_kernel<<<(CDq*NHq*GDq + 255)/256, 256, 0, stream>>>(gat_W, WCAT);
    gemm(stream, H, 0,0, CDq,0, WCAT, 0,0, NHq*GDq,0, WHALL, 0,0, NHq*GDq,
         nullptr, nullptr,0, nullptr,0, Bq*Nq, NHq*GDq, CDq, 1,1, 0);

    // ---- 3) head attentions ----
    gat_scores_kernel<<<(Bq*NHq*Nq + 255)/256, 256, 0, stream>>>(
        WHALL, NHq*GDq, gat_a, 2*GDq, SB1, SB2, NHq, GDq);
    {
        dim3 g(Nq, NHq, Bq);
        gat_att_kernel<<<g, 128, 0, stream>>>(SB1, SB2, adjacency, ATTB, NHq);
    }
    // hp = elu(att @ Wh) batched over (b, head); write straight into MULTI
    gemm(stream, ATTB, (long long)NHq*NN, NN, Nq, 0,
         WHALL, (long long)Nq*NHq*GDq, GDq, NHq*GDq, 0,
         MULTI, (long long)Nq*NHq*GDq, GDq, NHq*GDq,
         nullptr, nullptr,0, nullptr,0, Nq, GDq, Nq, Bq*NHq, NHq, 2);

    // ---- 4) output GAT ----
    gemm(stream, MULTI, 0,0, NHq*GDq,0, gatout_W, 0,0, CDq,0, WH2, 0,0, CDq,
         nullptr, nullptr,0, nullptr,0, Bq*Nq, CDq, NHq*GDq, 1,1, 0);
    gat_scores_kernel<<<(Bq*Nq + 255)/256, 256, 0, stream>>>(
        WH2, CDq, gatout_a, 2*CDq, SB1o, SB2o, 1, CDq);
    {
        dim3 g(Nq, 1, Bq);
        gat_att_kernel<<<g, 128, 0, stream>>>(SB1o, SB2o, adjacency, ATT2, 1);
    }
    gemm(stream, ATT2, NN,0, Nq,0, WH2, (long long)Nq*CDq,0, CDq,0,
         HP2, (long long)Nq*CDq,0, CDq,
         nullptr, nullptr,0, nullptr,0, Nq, CDq, Nq, Bq,1, 2);  // elu
    // av = leaky_relu(hp2 @ Wcomp + b)
    gemm(stream, HP2, 0,0, CDq,0, Wcomp_W, 0,0, LDq,0, AV, 0,0, LDq,
         Wcomp_b, nullptr,0, nullptr,0, Bq*Nq, LDq, CDq, 1,1, 1);

    // ---- 5) protein path ----
    gemm(stream, amino_emb, 0,0, 1024,0, prot_W, 0,0, PDq,0, CA, 0,0, PDq,
         prot_b, nullptr,0, nullptr,0, Bq*Mq, PDq, 1024, 1,1, 0);
    {
        int total = Bq * Mq * PDq, blk = 256, grd = (total + blk - 1)/blk;
        conv2d_kernel<<<grd, blk, 0, stream>>>(CA, conv_W + 0*KWq*KWq, conv_b + 0, CB);
        conv2d_kernel<<<grd, blk, 0, stream>>>(CB, conv_W + 1*KWq*KWq, conv_b + 1, CA);
        conv2d_kernel<<<grd, blk, 0, stream>>>(CA, conv_W + 2*KWq*KWq, conv_b + 2, CB);
    }
    gemm(stream, CB, 0,0, PDq,0, Wprot_W, 0,0, LDq,0, PV, 0,0, LDq,
         Wprot_b, nullptr,0, nullptr,0, Bq*Mq, LDq, PDq, 1,1, 1);

    // ---- 6) bidirectional attention, BIDAT=4 ----
    for (int i = 0; i < 4; ++i) {
        const long long w2 = (long long)LDq * LDq;
        gemm(stream, PV, 0,0, LDq,0, tp2c_W + i*w2, 0,0, LDq,0, TPC, 0,0, LDq,
             tp2c_b + i*LDq, nullptr,0, nullptr,0, Bq*Mq, LDq, LDq, 1,1, 3);
        gemm(stream, AV, 0,0, LDq,0, tc2p_W + i*w2, 0,0, LDq,0, TCP, 0,0, LDq,
             tc2p_b + i*LDq, nullptr,0, nullptr,0, Bq*Nq, LDq, LDq, 1,1, 3);
        gemm(stream, AV, 0,0, LDq,0, bhc_W + i*w2, 0,0, LDq,0, BHC, 0,0, LDq,
             bhc_b + i*LDq, nullptr,0, nullptr,0, Bq*Nq, LDq, LDq, 1,1, 3);
        gemm(stream, PV, 0,0, LDq,0, bhp_W + i*w2, 0,0, LDq,0, BHP, 0,0, LDq,
             bhp_b + i*LDq, nullptr,0, nullptr,0, Bq*Mq, LDq, LDq, 1,1, 3);
        gemm(stream, AV, 0,0, LDq,0, Uw + i*w2, 0,0, LDq,0, AVU, 0,0, LDq,
             nullptr, nullptr,0, nullptr,0, Bq*Nq, LDq, LDq, 1,1, 0);
        // A = tanh(avU @ pv^T) * atoms_mask[:,None] * amino_mask[None,:]
        gemm(stream, AVU, (long long)Nq*LDq,0, LDq,0,
             PV, (long long)Mq*LDq,0, LDq,1,
             ABUF, (long long)Nq*Mq,0, Mq,
             nullptr, atoms_mask, Nq, amino_mask, Mq,
             Nq, Mq, LDq, Bq,1, 3);
        // atoms_trans = A @ tanh(pv@tp2c)
        gemm(stream, ABUF, (long long)Nq*Mq,0, Mq,0,
             TPC, (long long)Mq*LDq,0, LDq,0,
             ATR, (long long)Nq*LDq,0, LDq,
             nullptr, nullptr,0, nullptr,0, Nq, LDq, Mq, Bq,1, 0);
        // amino_trans = A^T @ tanh(av@tc2p)
        gemm(stream, ABUF, (long long)Nq*Mq,0, Mq,1,
             TCP, (long long)Nq*LDq,0, LDq,0,
             AMT, (long long)Mq*LDq,0, LDq,
             nullptr, nullptr,0, nullptr,0, Mq, LDq, Nq, Bq,1, 0);
        // attention scores + masked softmax
        atoms_att_kernel<<<Bq, 128, 0, stream>>>(BHC, ATR, battc_W + i*2*LDq,
                                                 battc_b + i, atoms_mask, SCC);
        amino_att_kernel<<<Bq, 1024, 0, stream>>>(BHP, AMT, battp_W + i*2*LDq,
                                                  battp_b + i, amino_mask, SCP);
        // pooled features into concatenated buffers
        pool_kernel<<<Bq, LDq, 0, stream>>>(AV, SCC, CATCF, Nq, i*LDq, NHq*LDq);
        pool_kernel<<<Bq, LDq, 0, stream>>>(PV, SCP, CATPF, Mq, i*LDq, NHq*LDq);
    }

    // ---- 7) combine + head MLP ----
    gemm(stream, CATCF, 0,0, NHq*LDq,0, combc_W, 0,0, LDq,0, CF, 0,0, LDq,
         combc_b, nullptr,0, nullptr,0, Bq, LDq, NHq*LDq, 1,1, 0);
    gemm(stream, CATPF, 0,0, NHq*LDq,0, combp_W, 0,0, LDq,0, PF, 0,0, LDq,
         combp_b, nullptr,0, nullptr,0, Bq, LDq, NHq*LDq, 1,1, 0);
    build_v_kernel<<<Bq, 800, 0, stream>>>(CF, PF, fps, inv_Temp, Temp, V0);

    float* vin = V0; float* vout = V1;
    for (int j = 0; j < 3; ++j) {
        gemm(stream, vin, 0,0, VD,0, Wout_W + (long long)j*VD*VD, 0,0, VD,0,
             vout, 0,0, VD, Wout_b + j*VD,
             nullptr,0, nullptr,0, Bq, VD, VD, 1,1, 1);
        float* t = vin; vin = vout; vout = t;
    }
    out_kernel<<<1, Bq, 0, stream>>>(vin, out_W, out_b, out);
}